// VQ_53721450938442
// MI455X (gfx1250) — compile-verified
//
#include <hip/hip_runtime.h>
#include <hip/hip_bf16.h>
#include <math.h>

typedef __attribute__((ext_vector_type(2))) float v2f;
typedef __attribute__((ext_vector_type(8))) float v8f;

#define VQ_N 262144
#define VQ_D 512
#define VQ_K 512
#define DISC 0.9f

// LDS strides (floats). Chosen so 16 consecutive rows hit 16 distinct banks
// and row*stride stays 16B-aligned for float4 staging stores.
#define BSTR 36   // B tile stride (512 x 36)
#define ASTR 36   // A tile stride (64 x 36)
#define SSTR 524  // score tile stride (64 x 524); 8*524 % 64 == 32 -> no half-wave conflict

#define LDS_B_OFF   0
#define LDS_A_OFF   (512 * BSTR)                 // 18432
#define LDS_S_OFF   0                            // scores alias A+B (temporally separated)
#define LDS_CODES   (64 * SSTR)                  // 33536 (beyond score region)
#define LDS_FLOATS  (64 * SSTR + 64)             // 33600 floats = 134400 B

// ---------------- Kernel 1: prototypes, p2, zero sums/counts ----------------
__global__ __launch_bounds__(256) void vq_prep(
    const float* __restrict__ comb, const float* __restrict__ w,
    float* __restrict__ proto, float* __restrict__ p2,
    float* __restrict__ sums, float* __restrict__ counts) {
  const int k = blockIdx.x;
  const int t = threadIdx.x;
  float ss = 0.f;
  for (int d = t; d < VQ_D; d += 256) {
    float p = comb[k * VQ_D + d] / w[d];
    proto[k * VQ_D + d] = p;
    sums[k * VQ_D + d] = 0.f;
    ss += p * p;
  }
  __shared__ float red[256];
  red[t] = ss;
  __syncthreads();
  for (int s = 128; s > 0; s >>= 1) {
    if (t < s) red[t] += red[t + s];
    __syncthreads();
  }
  if (t == 0) {
    p2[k] = red[0];
    counts[k] = 0.f;
  }
}

// ---------------- Kernel 2: fp32 WMMA GEMM + argmax + outputs ----------------
__global__ __launch_bounds__(256) void vq_main(
    const float* __restrict__ inputs, const float* __restrict__ proto,
    const float* __restrict__ p2,
    float* __restrict__ onehot, float* __restrict__ closest,
    float* __restrict__ sums, float* __restrict__ counts) {
  extern __shared__ float lds[];
  float* Blds = lds + LDS_B_OFF;   // 512 x BSTR
  float* Alds = lds + LDS_A_OFF;   // 64  x ASTR
  float* Slds = lds + LDS_S_OFF;   // 64  x SSTR (aliases A/B after GEMM)
  int*   Codes = (int*)(lds + LDS_CODES);

  const int tid = threadIdx.x;
  const int lane = tid & 31;
  const int wave = tid >> 5;
  const int row0 = blockIdx.x * 64;

  const int wm = wave >> 2;  // 0..1 : rows [wm*32, wm*32+32)
  const int wn = wave & 3;   // 0..3 : cols [wn*128, wn*128+128)

  const int lhalf = lane >> 4;  // 0/1: which K-pair / M+8 half
  const int l15 = lane & 15;

  v8f acc[2][8];
#pragma unroll
  for (int i = 0; i < 2; i++)
#pragma unroll
    for (int j = 0; j < 8; j++) {
      v8f z = {};
      acc[i][j] = z;
    }

  for (int dbase = 0; dbase < VQ_D; dbase += 32) {
    // Stage B tile (512 x 32) : 4096 float4
    for (int idx = tid; idx < 4096; idx += 256) {
      int r = idx >> 3;
      int c4 = idx & 7;
      float4 v = ((const float4*)(proto + (size_t)r * VQ_D + dbase))[c4];
      *(float4*)(Blds + r * BSTR + c4 * 4) = v;
    }
    // Stage A tile (64 x 32) : 512 float4
    for (int idx = tid; idx < 512; idx += 256) {
      int r = idx >> 3;
      int c4 = idx & 7;
      float4 v = ((const float4*)(inputs + (size_t)(row0 + r) * VQ_D + dbase))[c4];
      *(float4*)(Alds + r * ASTR + c4 * 4) = v;
    }
    __syncthreads();

#pragma unroll
    for (int ks = 0; ks < 32; ks += 4) {
      // A fragments: lane holds A[m][ks + lhalf*2 + {0,1}]
      v2f a[2];
#pragma unroll
      for (int mt = 0; mt < 2; mt++) {
        int m = wm * 32 + mt * 16 + l15;
        a[mt] = *(const v2f*)(Alds + m * ASTR + ks + lhalf * 2);
      }
      // B fragments: lane holds P[n][ks + lhalf*2 + {0,1}]  (B = P^T)
      v2f b[8];
#pragma unroll
      for (int nt = 0; nt < 8; nt++) {
        int n = wn * 128 + nt * 16 + l15;
        b[nt] = *(const v2f*)(Blds + n * BSTR + ks + lhalf * 2);
      }
#pragma unroll
      for (int mt = 0; mt < 2; mt++)
#pragma unroll
        for (int nt = 0; nt < 8; nt++)
          acc[mt][nt] = __builtin_amdgcn_wmma_f32_16x16x4_f32(
              false, a[mt], false, b[nt], (short)0, acc[mt][nt], false, false);
    }
    __syncthreads();
  }

  // Dump scores = p2[col] - 2*dot  (argmax-equivalent to the distance)
#pragma unroll
  for (int mt = 0; mt < 2; mt++) {
#pragma unroll
    for (int nt = 0; nt < 8; nt++) {
      int col = wn * 128 + nt * 16 + l15;
      float pp = p2[col];
      int rbase = wm * 32 + mt * 16 + lhalf * 8;
#pragma unroll
      for (int v = 0; v < 8; v++)
        Slds[(rbase + v) * SSTR + col] = pp - 2.0f * acc[mt][nt][v];
    }
  }
  __syncthreads();

  // Per-row argmax (first-max-index tie rule). Wave w owns rows [w*8, w*8+8).
  for (int rr = 0; rr < 8; rr++) {
    int r = wave * 8 + rr;
    float best = -INFINITY;
    int bidx = 0;
    for (int c = lane; c < VQ_K; c += 32) {
      float v = Slds[r * SSTR + c];
      if (v > best) { best = v; bidx = c; }
    }
#pragma unroll
    for (int off = 16; off > 0; off >>= 1) {
      float ov = __shfl_xor(best, off, 32);
      int oi = __shfl_xor(bidx, off, 32);
      if (ov > best || (ov == best && oi < bidx)) { best = ov; bidx = oi; }
    }
    if (lane == 0) Codes[r] = bidx;
  }
  __syncthreads();

  // one_hot + closest: 64 rows x 128 float4
  for (int idx = tid; idx < 64 * 128; idx += 256) {
    int r = idx >> 7;
    int c4 = idx & 127;
    int code = Codes[r];
    size_t orow = (size_t)(row0 + r) * 512;
    float4 oh = make_float4(0.f, 0.f, 0.f, 0.f);
    int cb = c4 * 4;
    if (code >= cb && code < cb + 4) ((float*)&oh)[code - cb] = 1.0f;
    ((float4*)(onehot + orow))[c4] = oh;
    float4 cl = ((const float4*)(proto + (size_t)code * VQ_D))[c4];
    ((float4*)(closest + orow))[c4] = cl;
  }

  // scatter-add input rows into sums[code]
  for (int idx = tid; idx < 64 * 512; idx += 256) {
    int r = idx >> 9;
    int d = idx & 511;
    int code = Codes[r];
    float v = inputs[(size_t)(row0 + r) * VQ_D + d];
    atomicAdd(&sums[(size_t)code * VQ_D + d], v);
  }
  if (tid < 64) atomicAdd(&counts[Codes[tid]], 1.0f);
}

// ---------------- Kernel 3: EMA finalize ----------------
__global__ __launch_bounds__(256) void vq_finalize(
    const float* __restrict__ comb, const float* __restrict__ w,
    const float* __restrict__ sums, const float* __restrict__ counts,
    float* __restrict__ new_w, float* __restrict__ new_comb) {
  const int k = blockIdx.x;
  const int t = threadIdx.x;
  for (int d = t; d < VQ_D; d += 256)
    new_comb[k * VQ_D + d] =
        DISC * comb[k * VQ_D + d] + (1.0f - DISC) * sums[k * VQ_D + d];
  if (k == 0)
    for (int d = t; d < VQ_D; d += 256)
      new_w[d] = DISC * w[d] + (1.0f - DISC) * counts[d];
}

extern "C" void kernel_launch(void* const* d_in, const int* in_sizes, int n_in,
                              void* d_out, int out_size, void* d_ws, size_t ws_size,
                              hipStream_t stream) {
  const float* inputs = (const float*)d_in[0];
  const float* comb = (const float*)d_in[1];
  const float* w = (const float*)d_in[2];

  float* out = (float*)d_out;
  float* onehot = out;                                        // N*K
  float* closest = out + (size_t)VQ_N * VQ_K;                 // N*D
  float* new_w = out + (size_t)VQ_N * VQ_K + (size_t)VQ_N * VQ_D;
  float* new_comb = new_w + VQ_D;

  float* ws = (float*)d_ws;
  float* proto = ws;                           // K*D
  float* p2 = proto + (size_t)VQ_K * VQ_D;     // K
  float* sums = p2 + VQ_K;                     // K*D
  float* counts = sums + (size_t)VQ_K * VQ_D;  // K

  vq_prep<<<VQ_K, 256, 0, stream>>>(comb, w, proto, p2, sums, counts);

  const size_t dynLds = (size_t)LDS_FLOATS * sizeof(float);  // ~131 KB
  vq_main<<<VQ_N / 64, 256, dynLds, stream>>>(inputs, proto, p2, onehot,
                                              closest, sums, counts);

  vq_finalize<<<VQ_K, 256, 0, stream>>>(comb, w, sums, counts, new_w, new_comb);
}